// VoxelFeatureEncoding_54597624266811
// MI455X (gfx1250) — compile-verified
//
#include <hip/hip_runtime.h>
#include <hip/hip_bf16.h>

typedef float v2f __attribute__((ext_vector_type(2)));
typedef float v8f __attribute__((ext_vector_type(8)));

#define VFE_T   35
#define VFE_C   4
#define VFE_H   64
#define VFE_EPS 1e-5f

// ws float layout:
//   [0:64)    channel sum of x
//   [64:128)  channel sum of x^2
//   [128:192) scale  = gamma * rsqrt(var+eps)
//   [192:256) shift  = (bias - mu)*scale + beta
__global__ void vfe_zero_stats(float* ws) {
    int i = threadIdx.x;
    if (i < 128) ws[i] = 0.f;
}

__device__ __forceinline__ v8f wmma4(v2f a, v2f b, v8f c) {
    // D(16x16,f32) = A(16x4,f32) x B(4x16,f32) + C
    return __builtin_amdgcn_wmma_f32_16x16x4_f32(
        /*neg_a=*/false, a, /*neg_b=*/false, b,
        /*c_mod=*/(short)0, c, /*reuse_a=*/false, /*reuse_b=*/false);
}

// Per-voxel masked mean over T rows (channels 0..2 needed), whole-wave reduction.
__device__ __forceinline__ void voxel_mean(const float* __restrict__ vox, int v,
                                           int lane, float& mx, float& my, float& mz) {
    const float4* vp = reinterpret_cast<const float4*>(vox) + (size_t)v * VFE_T;
    float4 a = vp[lane];                       // rows 0..31
    float sx = a.x, sy = a.y, sz = a.z;
    float cnt = ((a.x + a.y + a.z + a.w) != 0.f) ? 1.f : 0.f;
    if (lane < VFE_T - 32) {                   // rows 32..34
        float4 b2 = vp[32 + lane];
        sx += b2.x; sy += b2.y; sz += b2.z;
        cnt += ((b2.x + b2.y + b2.z + b2.w) != 0.f) ? 1.f : 0.f;
    }
#pragma unroll
    for (int m = 16; m >= 1; m >>= 1) {
        sx  += __shfl_xor(sx,  m, 32);
        sy  += __shfl_xor(sy,  m, 32);
        sz  += __shfl_xor(sz,  m, 32);
        cnt += __shfl_xor(cnt, m, 32);
    }
    mx = sx / cnt; my = sy / cnt; mz = sz / cnt;
}

// Build the two A-operands (K-chunks 0..3 and 4..7, k=7 zero padded) for row tile mt.
// A(16x4,f32) layout: lane l holds row M=l%16; VGPR j holds K = j + 2*(l/16).
// half==0 -> a0={aug0,aug1}, a1={aug4,aug5}; half==1 -> a0={aug2,aug3}, a1={aug6,0}
__device__ __forceinline__ void load_A(const float* __restrict__ vox, int v, int mt,
                                       int nlo, int half, float mx, float my, float mz,
                                       v2f& a0, v2f& a1) {
    int row = mt * 16 + nlo;
    bool ok = row < VFE_T;
    float4 p = make_float4(0.f, 0.f, 0.f, 0.f);
    if (ok) {
        const float4* vp = reinterpret_cast<const float4*>(vox) + (size_t)v * VFE_T;
        p = vp[row];
    }
    float r0 = ok ? p.x - mx : 0.f;
    float r1 = ok ? p.y - my : 0.f;
    float r2 = ok ? p.z - mz : 0.f;
    a0[0] = half ? p.z : p.x;
    a0[1] = half ? p.w : p.y;
    a1[0] = half ? r2  : r0;
    a1[1] = half ? 0.f : r1;
}

// B(4x16,f32) = W.T tile: lane l holds column N=l%16; VGPR j holds K = j + 2*(l/16).
__device__ __forceinline__ void load_B(const float* __restrict__ W, int nlo, int half,
                                       v2f B[2][4]) {
#pragma unroll
    for (int nt = 0; nt < 4; ++nt) {
        int n = nt * 16 + nlo;
#pragma unroll
        for (int kc = 0; kc < 2; ++kc) {
            int k0 = kc * 4 + 2 * half;
            B[kc][nt][0] = (k0     < 7) ? W[n * 7 + k0]     : 0.f;
            B[kc][nt][1] = (k0 + 1 < 7) ? W[n * 7 + k0 + 1] : 0.f;
        }
    }
}

// Pass 1: accumulate per-channel sum / sumsq of x = aug @ W.T + b over all N*T rows.
__global__ void __launch_bounds__(256)
vfe_stats(const float* __restrict__ vox, const float* __restrict__ W,
          const float* __restrict__ bias, float* __restrict__ ws, int Nvox) {
    int lane = threadIdx.x & 31;
    int half = lane >> 4;
    int nlo  = lane & 15;
    int wave   = blockIdx.x * (blockDim.x >> 5) + (threadIdx.x >> 5);
    int nwaves = gridDim.x * (blockDim.x >> 5);

    v2f B[2][4];
    load_B(W, nlo, half, B);
    float bb[4];
#pragma unroll
    for (int nt = 0; nt < 4; ++nt) bb[nt] = bias[nt * 16 + nlo];

    float s[4]  = {0.f, 0.f, 0.f, 0.f};
    float s2[4] = {0.f, 0.f, 0.f, 0.f};

    for (int v = wave; v < Nvox; v += nwaves) {
        float mx, my, mz;
        voxel_mean(vox, v, lane, mx, my, mz);
#pragma unroll
        for (int mt = 0; mt < 3; ++mt) {
            v2f a0, a1;
            load_A(vox, v, mt, nlo, half, mx, my, mz, a0, a1);
#pragma unroll
            for (int nt = 0; nt < 4; ++nt) {
                v8f acc = {0.f, 0.f, 0.f, 0.f, 0.f, 0.f, 0.f, 0.f};
                acc = wmma4(a0, B[0][nt], acc);
                acc = wmma4(a1, B[1][nt], acc);
#pragma unroll
                for (int j = 0; j < 8; ++j) {
                    int row = mt * 16 + j + 8 * half;   // D: M = j + 8*(lane/16)
                    if (row < VFE_T) {
                        float x = acc[j] + bb[nt];
                        s[nt]  += x;
                        s2[nt] += x * x;
                    }
                }
            }
        }
    }
#pragma unroll
    for (int nt = 0; nt < 4; ++nt) {
        s[nt]  += __shfl_xor(s[nt],  16, 32);
        s2[nt] += __shfl_xor(s2[nt], 16, 32);
        if (half == 0) {
            atomicAdd(&ws[     nt * 16 + nlo], s[nt]);
            atomicAdd(&ws[64 + nt * 16 + nlo], s2[nt]);
        }
    }
}

// Pass 2: fold BN stats + bias into per-channel scale/shift.
__global__ void vfe_finalize(const float* __restrict__ bias,
                             const float* __restrict__ gamma,
                             const float* __restrict__ beta,
                             float* ws, float invM) {
    int c = threadIdx.x;
    if (c < VFE_H) {
        float mu  = ws[c] * invM;
        float var = ws[64 + c] * invM - mu * mu;
        float sc  = gamma[c] * rsqrtf(var + VFE_EPS);
        ws[128 + c] = sc;
        ws[192 + c] = (bias[c] - mu) * sc + beta[c];
    }
}

// Pass 3: recompute x, apply BN+ReLU, per-voxel max over T, write [N,T,128] output.
__global__ void __launch_bounds__(256)
vfe_output(const float* __restrict__ vox, const float* __restrict__ W,
           const float* __restrict__ ws, float* __restrict__ out, int Nvox) {
    int lane = threadIdx.x & 31;
    int half = lane >> 4;
    int nlo  = lane & 15;
    int wave   = blockIdx.x * (blockDim.x >> 5) + (threadIdx.x >> 5);
    int nwaves = gridDim.x * (blockDim.x >> 5);

    v2f B[2][4];
    load_B(W, nlo, half, B);
    float sc[4], sh[4];
#pragma unroll
    for (int nt = 0; nt < 4; ++nt) {
        sc[nt] = ws[128 + nt * 16 + nlo];
        sh[nt] = ws[192 + nt * 16 + nlo];
    }

    for (int v = wave; v < Nvox; v += nwaves) {
        float mx, my, mz;
        voxel_mean(vox, v, lane, mx, my, mz);
        float rmax[4] = {0.f, 0.f, 0.f, 0.f};   // ReLU output >= 0
        float* outv = out + (size_t)v * VFE_T * (2 * VFE_H);
#pragma unroll
        for (int mt = 0; mt < 3; ++mt) {
            v2f a0, a1;
            load_A(vox, v, mt, nlo, half, mx, my, mz, a0, a1);
#pragma unroll
            for (int nt = 0; nt < 4; ++nt) {
                v8f acc = {0.f, 0.f, 0.f, 0.f, 0.f, 0.f, 0.f, 0.f};
                acc = wmma4(a0, B[0][nt], acc);
                acc = wmma4(a1, B[1][nt], acc);
#pragma unroll
                for (int j = 0; j < 8; ++j) {
                    int row = mt * 16 + j + 8 * half;
                    if (row < VFE_T) {
                        float y = fmaf(acc[j], sc[nt], sh[nt]);   // bias folded in shift
                        y = fmaxf(y, 0.f);
                        outv[row * 128 + nt * 16 + nlo] = y;
                        rmax[nt] = fmaxf(rmax[nt], y);
                    }
                }
            }
        }
        int t0 = half ? 17 : 0;
        int t1 = half ? VFE_T : 17;
#pragma unroll
        for (int nt = 0; nt < 4; ++nt) {
            float m = fmaxf(rmax[nt], __shfl_xor(rmax[nt], 16, 32));
            for (int t = t0; t < t1; ++t)
                outv[t * 128 + 64 + nt * 16 + nlo] = m;
        }
    }
}

extern "C" void kernel_launch(void* const* d_in, const int* in_sizes, int n_in,
                              void* d_out, int out_size, void* d_ws, size_t ws_size,
                              hipStream_t stream) {
    const float* vox   = (const float*)d_in[0];
    const float* W     = (const float*)d_in[1];
    const float* bias  = (const float*)d_in[2];
    const float* gamma = (const float*)d_in[3];
    const float* beta  = (const float*)d_in[4];
    float* out = (float*)d_out;
    float* ws  = (float*)d_ws;

    int Nvox = in_sizes[0] / (VFE_T * VFE_C);   // 20000

    vfe_zero_stats<<<1, 128, 0, stream>>>(ws);

    // stats: 256 blocks x 8 waves = 2048 waves, ~10 voxels each
    vfe_stats<<<256, 256, 0, stream>>>(vox, W, bias, ws, Nvox);

    float invM = 1.0f / (float)((long long)Nvox * VFE_T);
    vfe_finalize<<<1, 64, 0, stream>>>(bias, gamma, beta, ws, invM);

    // output: one wave per voxel
    int blocks3 = (Nvox + 7) / 8;
    vfe_output<<<blocks3, 256, 0, stream>>>(vox, W, ws, out, Nvox);
}